// NemotronHTopkRouter_57647051047637
// MI455X (gfx1250) — compile-verified
//
#include <hip/hip_runtime.h>
#include <hip/hip_bf16.h>
#include <float.h>

// ---------------------------------------------------------------------------
// NemotronH / DeepSeek-V3 style MoE router for MI455X (gfx1250, wave32)
//   router GEMM in fp32 via V_WMMA_F32_16X16X4_F32, routing fused on-chip.
//   M-blocked (32 tokens/block) to halve weight-matrix L2 traffic and give
//   1.0 loads/WMMA issue ratio (A and B fragments each reused twice).
// ---------------------------------------------------------------------------

typedef float v2f __attribute__((ext_vector_type(2)));
typedef float v8f __attribute__((ext_vector_type(8)));

#define RT_TOKENS   16384
#define RT_HIDDEN   4096
#define RT_NEXP     256
#define RT_NGROUP   8
#define RT_TOPKG    4
#define RT_TOPK     8
#define RT_SCALE    2.5f
#define RT_MB       32          // tokens per block

__global__ __launch_bounds__(256)
void nemotron_router_kernel(const float* __restrict__ X,     // [T, H]
                            const float* __restrict__ Wt,    // [E, H]
                            const float* __restrict__ bias,  // [E]
                            int*   __restrict__ out_idx,     // [T, 8]
                            float* __restrict__ out_w)       // [T, 8]
{
    __shared__ float lds_logits[RT_MB * RT_NEXP];   // 32 KB: 32 tokens x 256 experts

    const int lane = threadIdx.x & 31;
    const int wave = threadIdx.x >> 5;
    const int row0 = blockIdx.x * RT_MB;            // 32 tokens per block

    // ======================= GEMM phase (fp32 WMMA) ========================
    // A fragment (16x4): lanes 0-15 rows M=0..15 hold K={k,k+1},
    //                    lanes 16-31 hold K={k+2,k+3}   (v0=K, v1=K+1)
    // B fragment (4x16): symmetric (N across lanes, K split across halves)
    const int mrow0 = row0 + (lane & 15);           // row tile 0
    const int mrow1 = mrow0 + 16;                   // row tile 1
    const int koff  = (lane >> 4) << 1;             // 0 or 2
    const int n0    = wave * 32;                    // this wave's 2 expert tiles
    const int col0  = n0 + (lane & 15);
    const int col1  = col0 + 16;

    const float* aptr0 = X  + (size_t)mrow0 * RT_HIDDEN + koff;
    const float* aptr1 = X  + (size_t)mrow1 * RT_HIDDEN + koff;
    const float* b0ptr = Wt + (size_t)col0  * RT_HIDDEN + koff;
    const float* b1ptr = Wt + (size_t)col1  * RT_HIDDEN + koff;

    v8f acc00 = {};   // (mtile0, ntile0)
    v8f acc01 = {};   // (mtile0, ntile1)
    v8f acc10 = {};   // (mtile1, ntile0)
    v8f acc11 = {};   // (mtile1, ntile1)
#pragma unroll 4
    for (int k = 0; k < RT_HIDDEN; k += 4) {
        v2f a0 = *(const v2f*)(aptr0 + k);
        v2f a1 = *(const v2f*)(aptr1 + k);
        v2f b0 = *(const v2f*)(b0ptr + k);
        v2f b1 = *(const v2f*)(b1ptr + k);
        acc00 = __builtin_amdgcn_wmma_f32_16x16x4_f32(false, a0, false, b0,
                                                      (short)0, acc00, false, false);
        acc01 = __builtin_amdgcn_wmma_f32_16x16x4_f32(false, a0, false, b1,
                                                      (short)0, acc01, false, false);
        acc10 = __builtin_amdgcn_wmma_f32_16x16x4_f32(false, a1, false, b0,
                                                      (short)0, acc10, false, false);
        acc11 = __builtin_amdgcn_wmma_f32_16x16x4_f32(false, a1, false, b1,
                                                      (short)0, acc11, false, false);
    }

    // C/D layout: VGPR r -> row (r + 8*(lane>=16)), col = lane&15
    {
        const int crow = (lane >> 4) << 3;          // 0 or 8
        const int ccol = lane & 15;
#pragma unroll
        for (int r = 0; r < 8; ++r) {
            const int lr0 = (crow + r) * RT_NEXP;           // mtile0 rows
            const int lr1 = (16 + crow + r) * RT_NEXP;      // mtile1 rows
            lds_logits[lr0 + n0 + ccol]      = acc00[r];
            lds_logits[lr0 + n0 + 16 + ccol] = acc01[r];
            lds_logits[lr1 + n0 + ccol]      = acc10[r];
            lds_logits[lr1 + n0 + 16 + ccol] = acc11[r];
        }
    }
    __syncthreads();

    // ========================= Routing phase ===============================
    // One wave per token, each wave processes 4 of the 32 tokens.
    // Lane l owns experts [8l, 8l+8); group g (32 experts) = lanes 4g..4g+3.
    const int e0 = lane * RT_TOPK;                  // 8 experts per lane
    float bl[8];
#pragma unroll
    for (int j = 0; j < 8; ++j) bl[j] = bias[e0 + j];

    const float NEG = -FLT_MAX;                     // jnp.finfo(float32).min

    for (int ti = 0; ti < RT_MB / 8; ++ti) {
        const int t = wave * (RT_MB / 8) + ti;      // local token 0..31
        const float* lrow = &lds_logits[t * RT_NEXP + e0];

        float s[8], bsc[8];
#pragma unroll
        for (int j = 0; j < 8; ++j) {
            float x  = lrow[j];
            float sg = 1.0f / (1.0f + __expf(-x));  // sigmoid
            s[j]   = sg;
            bsc[j] = sg + bl[j];
        }

        // ---- local top-2 of this lane's 8 biased scores ----
        float m1 = NEG, m2 = NEG;
#pragma unroll
        for (int j = 0; j < 8; ++j) {
            float v = bsc[j];
            if (v > m1)      { m2 = m1; m1 = v; }
            else if (v > m2) { m2 = v; }
        }
        // ---- merge top-2 across the 4 lanes of this group (butterfly) ----
#pragma unroll
        for (int off = 1; off <= 2; off <<= 1) {
            float o1 = __shfl_xor(m1, off, 32);
            float o2 = __shfl_xor(m2, off, 32);
            if (o1 > m1) { m2 = fmaxf(m1, o2); m1 = o1; }
            else         { m2 = fmaxf(m2, o1); }
        }
        const float gscore = m1 + m2;               // group score (all 4 lanes)

        // ---- gather all 8 group scores, rank my group, top-4 groups ----
        float gsv[8];
#pragma unroll
        for (int g = 0; g < 8; ++g) gsv[g] = __shfl(gscore, g << 2, 32);
        const int gm = lane >> 2;
        int rank = 0;
#pragma unroll
        for (int g = 0; g < 8; ++g)
            rank += (gsv[g] > gscore) || (gsv[g] == gscore && g < gm);
        const bool keep = (rank < RT_TOPKG);

        float bm[8];
#pragma unroll
        for (int j = 0; j < 8; ++j) bm[j] = keep ? bsc[j] : NEG;

        // ---- iterative wave-wide top-8 (value desc, index asc on ties) ----
        float wsum = 0.0f;
        int   idxs[RT_TOPK];
        float wts[RT_TOPK];
#pragma unroll
        for (int r = 0; r < RT_TOPK; ++r) {
            float v  = bm[0];
            float sv = s[0];
            int   ix = e0;
#pragma unroll
            for (int j = 1; j < 8; ++j)
                if (bm[j] > v) { v = bm[j]; sv = s[j]; ix = e0 + j; }
#pragma unroll
            for (int off = 16; off > 0; off >>= 1) {
                float ov = __shfl_xor(v,  off, 32);
                float os = __shfl_xor(sv, off, 32);
                int   oi = __shfl_xor(ix, off, 32);
                if (ov > v || (ov == v && oi < ix)) { v = ov; sv = os; ix = oi; }
            }
            // owner lane retires the winner
#pragma unroll
            for (int j = 0; j < 8; ++j)
                if (ix == e0 + j) bm[j] = NEG;
            idxs[r] = ix;
            wts[r]  = sv;                           // UNbiased sigmoid score
            wsum   += sv;
        }

        const float inv = RT_SCALE / (wsum + 1e-20f);
        if (lane == 0) {
            const size_t base = (size_t)(row0 + t) * RT_TOPK;
            int4 i0 = make_int4(idxs[0], idxs[1], idxs[2], idxs[3]);
            int4 i1 = make_int4(idxs[4], idxs[5], idxs[6], idxs[7]);
            float4 w0 = make_float4(wts[0] * inv, wts[1] * inv, wts[2] * inv, wts[3] * inv);
            float4 w1 = make_float4(wts[4] * inv, wts[5] * inv, wts[6] * inv, wts[7] * inv);
            *(int4*)  (out_idx + base)     = i0;
            *(int4*)  (out_idx + base + 4) = i1;
            *(float4*)(out_w   + base)     = w0;
            *(float4*)(out_w   + base + 4) = w1;
        }
    }
}

extern "C" void kernel_launch(void* const* d_in, const int* in_sizes, int n_in,
                              void* d_out, int out_size, void* d_ws, size_t ws_size,
                              hipStream_t stream) {
    const float* X    = (const float*)d_in[0];   // hidden_states [16384, 4096]
    const float* Wt   = (const float*)d_in[1];   // weight        [256, 4096]
    const float* bias = (const float*)d_in[2];   // bias          [256]

    // Tuple output, concatenated flat: topk_indices (int32) then topk_weights (f32)
    int*   out_idx = (int*)d_out;
    float* out_w   = (float*)d_out + (size_t)RT_TOKENS * RT_TOPK;

    dim3 grid(RT_TOKENS / RT_MB);
    dim3 block(256);
    hipLaunchKernelGGL(nemotron_router_kernel, grid, block, 0, stream,
                       X, Wt, bias, out_idx, out_w);
}